// Position_linear_34102040330537
// MI455X (gfx1250) — compile-verified
//
#include <hip/hip_runtime.h>
#include <hip/hip_bf16.h>

typedef __attribute__((ext_vector_type(2))) float v2f;
typedef __attribute__((ext_vector_type(8))) float v8f;
typedef __attribute__((ext_vector_type(4))) int   v4i;

#define SEQ     41
#define FEAT    128
#define KDIM    384            // 3 * FEAT
#define NDIM    128
#define ROWSTR  (SEQ * FEAT)   // 5248 floats between batch rows
#define MTILES  256            // 4096 / 16
#define LDS_STRIDE 136         // 128 + 8 dword pad: half-wave bank sets disjoint (2*136 % 64 == 16)
#define LDS_FLOATS (KDIM * LDS_STRIDE)       // 52224 floats
#define LDS_BYTES  (LDS_FLOATS * 4)          // 208896 B (fits 320KB/WGP)

#if defined(__has_builtin)
#  if __has_builtin(__builtin_amdgcn_global_load_async_to_lds_b128)
#    define HAVE_ASYNC_LDS 1
#  endif
#endif

// One wave computes a 16(M) x 128(N) output tile for one sequence position s.
// All 16 waves of a block share s; W_s (384x128 f32) is staged once into LDS
// (async-to-LDS DMA when available), then the K-loop runs 8x
// V_WMMA_F32_16X16X4_F32 per 4-wide K-chunk with A from global and B from LDS.
__global__ __launch_bounds__(512)
void Position_linear_34102040330537_kernel(const float* __restrict__ x,
                                           const float* __restrict__ W,
                                           const float* __restrict__ bias,
                                           float* __restrict__ out)
{
    extern __shared__ float wlds[];  // [KDIM][LDS_STRIDE]

    const int tid   = threadIdx.x;
    const int widx  = tid >> 5;                    // wave in block, 0..15
    const int lane  = tid & 31;
    const int s     = blockIdx.x >> 4;             // block-uniform sequence position
    const int mtile = ((blockIdx.x & 15) << 4) + widx;
    const int m_base = mtile << 4;                 // * 16

    // ---- Stage W_s into LDS with row padding (row k at dword k*LDS_STRIDE) ----
    {
        const float* wsrc = W + (long)s * (KDIM * NDIM);
        // 12288 b128 chunks (384 rows * 32 chunks), 512 threads -> 24 exact iters
        #pragma unroll 4
        for (int c = tid; c < (KDIM * NDIM) / 4; c += 512) {
            const int row  = c >> 5;        // / 32
            const int col4 = (c & 31) << 2; // * 4
#ifdef HAVE_ASYNC_LDS
            __builtin_amdgcn_global_load_async_to_lds_b128(
                (v4i*)(wsrc + row * NDIM + col4),
                (v4i*)(wlds + row * LDS_STRIDE + col4),
                0, 0);
#else
            const float4 v = *(const float4*)(wsrc + row * NDIM + col4);
            *(float4*)(wlds + row * LDS_STRIDE + col4) = v;
#endif
        }
#ifdef HAVE_ASYNC_LDS
#  if __has_builtin(__builtin_amdgcn_s_wait_asynccnt)
        __builtin_amdgcn_s_wait_asynccnt(0);
#  else
        asm volatile("s_wait_asynccnt 0x0" ::: "memory");
#  endif
#endif
        __syncthreads();
    }

    const int lhalf = lane >> 4;          // 0: lanes 0-15, 1: lanes 16-31
    const int lmod  = lane & 15;

    // A-fragment (16x4 f32): lane holds A[m = lmod][k0 + 2*lhalf + {0,1}]
    const long arow = (long)(m_base + lmod) * ROWSTR + (long)(s - 1) * FEAT;

    // B-fragment (4x16 f32) from LDS: lane holds W_s[k0 + 2*lhalf + {0,1}][nt*16 + lmod]
    const int bbase = 2 * lhalf * LDS_STRIDE + lmod;

    // Zero padding at sequence edges == skip out-of-range K blocks.
    const int k_lo = (s == 0)       ? FEAT     : 0;
    const int k_hi = (s == SEQ - 1) ? 2 * FEAT : KDIM;

    v8f acc[8] = {}; // 8 N-tiles of 16 cols -> full N = 128

    for (int k0 = k_lo; k0 < k_hi; k0 += 4) {
        const v2f a = *(const v2f*)(x + arow + k0 + 2 * lhalf);
        const float* wp = wlds + k0 * LDS_STRIDE + bbase;
        #pragma unroll
        for (int nt = 0; nt < 8; ++nt) {
            v2f b;
            b.x = wp[nt * 16];               // W_s[k][n]
            b.y = wp[nt * 16 + LDS_STRIDE];  // W_s[k+1][n]
            // (neg_a, A, neg_b, B, c_mod, C, reuse_a, reuse_b)
            acc[nt] = __builtin_amdgcn_wmma_f32_16x16x4_f32(
                false, a, false, b, (short)0, acc[nt], false, false);
        }
    }

    // C layout: VGPR r, lane l -> m = r + 8*lhalf, n = nt*16 + lmod.
    #pragma unroll
    for (int nt = 0; nt < 8; ++nt) {
        const float bv = bias[s * NDIM + nt * 16 + lmod];
        const long obase = (long)(m_base + 8 * lhalf) * ROWSTR
                         + (long)s * FEAT + nt * 16 + lmod;
        #pragma unroll
        for (int r = 0; r < 8; ++r) {
            float v = acc[nt][r] + bv;
            v = v > 0.f ? v : 0.f;
            out[obase + (long)r * ROWSTR] = v;
        }
    }
}

extern "C" void kernel_launch(void* const* d_in, const int* in_sizes, int n_in,
                              void* d_out, int out_size, void* d_ws, size_t ws_size,
                              hipStream_t stream) {
    const float* x    = (const float*)d_in[0]; // (4096, 41, 128)
    const float* W    = (const float*)d_in[1]; // (41, 384, 128)
    const float* bias = (const float*)d_in[2]; // (41, 128)
    float* out        = (float*)d_out;         // (4096, 41, 128)

    // 41 s-positions * 16 blocks each (16 waves x 16 M-tiles per block).
    const int blocks = SEQ * 16;               // 656, exact cover
    Position_linear_34102040330537_kernel<<<blocks, 512, LDS_BYTES, stream>>>(x, W, bias, out);
}